// SiameseGATArm_54417235640477
// MI455X (gfx1250) — compile-verified
//
#include <hip/hip_runtime.h>
#include <hip/hip_bf16.h>
#include <math.h>

// ---------------------------------------------------------------------------
// SiameseGATArm for MI455X (gfx1250, wave32).
// GEMMs: V_WMMA_F32_16X16X4_F32 on a zero-padded packed A. Per K-chunk the
// wave batch-prefetches its whole A strip into registers (16 x b64 loads);
// the B tile is stored K-contiguous in LDS so every B fragment is a single
// aligned ds_load_b64 straight into the WMMA operand pair (no shuffle movs).
// Edge softmax/aggregation uses HW float atomics; h + accumulator fit in the
// 192MB L2 so the scatter phase is L2-resident.
// ---------------------------------------------------------------------------

typedef __attribute__((ext_vector_type(2))) float v2f;
typedef __attribute__((ext_vector_type(8))) float v8f;

#define NEG_SLOPE 0.2f
#define BN_EPS 1e-5f

__device__ __forceinline__ float atomicMaxF(float* addr, float val) {
  // IEEE ordering trick: int-max for >=0, uint-min for <0. Works with -1e30 init.
  if (val >= 0.f)
    return __int_as_float(atomicMax((int*)addr, __float_as_int(val)));
  else
    return __uint_as_float(atomicMin((unsigned int*)addr, __float_as_uint(val)));
}

// ---------------------------------------------------------------------------
// Fill
// ---------------------------------------------------------------------------
__global__ void fill_kernel(float* p, float v, size_t n) {
  size_t i = (size_t)blockIdx.x * blockDim.x + threadIdx.x;
  size_t stride = (size_t)gridDim.x * blockDim.x;
  for (; i < n; i += stride) p[i] = v;
}

// ---------------------------------------------------------------------------
// Pack A (+ optional residual A2) into zero-padded Apad[Mpad, Kpad].
// All GEMM bounds handling happens here, once, outside the WMMA hot loop.
// ---------------------------------------------------------------------------
__global__ void pack_kernel(const float* __restrict__ A,
                            const float* __restrict__ A2,
                            float* __restrict__ Apad, int M, int K, int Kpad,
                            size_t total) {
  size_t i = (size_t)blockIdx.x * blockDim.x + threadIdx.x;
  size_t stride = (size_t)gridDim.x * blockDim.x;
  for (; i < total; i += stride) {
    int row = (int)(i / Kpad);
    int k = (int)(i % Kpad);
    float v = 0.f;
    if (row < M && k < K) {
      v = A[(size_t)row * K + k];
      if (A2) v += A2[(size_t)row * K + k];
    }
    Apad[i] = v;
  }
}

// ---------------------------------------------------------------------------
// WMMA GEMM: H[M,N] = Apad @ W.  Apad:[Mpad,Kpad] zero-padded row-major,
// W:[K,N] row-major. Block 256 thr (8 waves), tile 128(M) x 64(N); wave w
// owns rows [w*16, w*16+16) as 4 16x16 f32 accumulators.
// Kpad % 64 == 0, Mpad % 128 == 0 -> divergence-free inner loop.
// B tile in LDS stored transposed, Bs[n][k], row stride 68:
//   fragment read = one ds_load_b64 of {B[k],B[k+1]}; banks (4i+kB) vs
//   (4i+kB+2) are disjoint across the two half-waves -> conflict-free.
// ---------------------------------------------------------------------------
#define KCHUNK 64
#define BN_STRIDE 68

__global__ __launch_bounds__(256) void gemm_wmma_kernel(
    const float* __restrict__ Apad, const float* __restrict__ W,
    float* __restrict__ H, int M, int Kpad, int K, int N) {
  __shared__ float Bs[64 * BN_STRIDE];  // 17KB, [n][k] layout

  const int wave = threadIdx.x >> 5;
  const int lane = threadIdx.x & 31;
  const int row0 = blockIdx.x * 128 + wave * 16;
  const int col0 = blockIdx.y * 64;

  v8f acc[4] = {};  // 4 x (16x16) f32 accumulators covering 64 columns

  // A-fragment addressing (ISA 7.12.2, 32-bit A 16x4):
  //   lane l: M = l&15, K-pair base = 2*(l>>4)   (pair offset baked into Arow)
  const float* Arow =
      Apad + (size_t)(row0 + (lane & 15)) * Kpad + ((lane >> 4) << 1);
  const int nsub = lane & 15;
  const int kLanePair = (lane >> 4) << 1;

  for (int kc = 0; kc < Kpad; kc += KCHUNK) {
    // Stage W[kc..kc+63][col0..col0+63] into LDS transposed: Bs[n][k].
    // Global reads stay coalesced (consecutive t -> consecutive n).
    for (int i = threadIdx.x; i < KCHUNK * 64; i += 256) {
      int kk = i >> 6, nn = i & 63;
      int k = kc + kk;
      Bs[nn * BN_STRIDE + kk] = (k < K) ? W[(size_t)k * N + col0 + nn] : 0.f;
    }
    __syncthreads();

    // Batch-prefetch this wave's entire A strip for the chunk: 16 x b64,
    // issued back-to-back so loadcnt drains behind the first LDS reads.
    v2f areg[16];
#pragma unroll
    for (int j = 0; j < 16; ++j)
      areg[j] = *(const v2f*)(Arow + kc + 4 * j);  // aligned, unconditional

#pragma unroll
    for (int k4 = 0; k4 < KCHUNK; k4 += 4) {
      const v2f a = areg[k4 >> 2];
#pragma unroll
      for (int s = 0; s < 4; ++s) {
        // {B[k], B[k+1]} for this lane's column: single aligned ds_load_b64.
        const v2f b = *(const v2f*)(
            &Bs[(s * 16 + nsub) * BN_STRIDE + k4 + kLanePair]);
        acc[s] = __builtin_amdgcn_wmma_f32_16x16x4_f32(
            false, a, false, b, (short)0, acc[s], false, false);
      }
    }
    __syncthreads();
  }

  // C/D layout: lane l, VGPR r -> row = r + 8*(l>>4), col = l&15
  const int rr = (lane >> 4) * 8;
  const int cc = lane & 15;
#pragma unroll
  for (int s = 0; s < 4; ++s) {
#pragma unroll
    for (int r = 0; r < 8; ++r) {
      const int mrow = row0 + rr + r;
      if (mrow < M) H[(size_t)mrow * N + col0 + s * 16 + cc] = acc[s][r];
    }
  }
}

// ---------------------------------------------------------------------------
// alpha_src / alpha_dst: per (node, head) dot of h[n,h,:] with a[h,:]
// One 256-thread block per node; D = heads*C <= 256.
// ---------------------------------------------------------------------------
__global__ __launch_bounds__(256) void alpha_kernel(
    const float* __restrict__ Hm, const float* __restrict__ a_s,
    const float* __restrict__ a_d, float* __restrict__ as_out,
    float* __restrict__ ad_out, int heads, int C) {
  __shared__ float ss[256];
  __shared__ float sd[256];
  const int node = blockIdx.x;
  const int D = heads * C;
  const int t = threadIdx.x;
  float hv = (t < D) ? Hm[(size_t)node * D + t] : 0.f;
  ss[t] = (t < D) ? hv * a_s[t] : 0.f;
  sd[t] = (t < D) ? hv * a_d[t] : 0.f;
  __syncthreads();
  if (t < heads) {
    float s = 0.f, d = 0.f;
    for (int c = 0; c < C; ++c) {
      s += ss[t * C + c];
      d += sd[t * C + c];
    }
    as_out[(size_t)node * heads + t] = s;
    ad_out[(size_t)node * heads + t] = d;
  }
}

// ---------------------------------------------------------------------------
// Edge pass 1: e = leaky_relu(as[src]+ad[dst]); stash to ework; segment max.
// Edges e < E come from edge_index; e >= E are self-loops (src=dst=e-E).
// ---------------------------------------------------------------------------
__global__ void edge_max_kernel(const int* __restrict__ ei, int E, int Nn,
                                int heads, const float* __restrict__ as,
                                const float* __restrict__ ad,
                                float* __restrict__ ework,
                                float* __restrict__ mmax) {
  int e = blockIdx.x * blockDim.x + threadIdx.x;
  if (e >= E + Nn) return;
  int s = (e < E) ? ei[e] : (e - E);
  int d = (e < E) ? ei[E + e] : (e - E);
  for (int h = 0; h < heads; ++h) {
    float v = as[(size_t)s * heads + h] + ad[(size_t)d * heads + h];
    v = (v > 0.f) ? v : NEG_SLOPE * v;
    ework[(size_t)e * heads + h] = v;
    atomicMaxF(&mmax[(size_t)d * heads + h], v);
  }
}

// Edge pass 2: exp(e - max[dst]); stash; segment sum.
__global__ void edge_sum_kernel(const int* __restrict__ ei, int E, int Nn,
                                int heads, const float* __restrict__ mmax,
                                float* __restrict__ ework,
                                float* __restrict__ denom) {
  int e = blockIdx.x * blockDim.x + threadIdx.x;
  if (e >= E + Nn) return;
  int d = (e < E) ? ei[E + e] : (e - E);
  for (int h = 0; h < heads; ++h) {
    float v = __expf(ework[(size_t)e * heads + h] - mmax[(size_t)d * heads + h]);
    ework[(size_t)e * heads + h] = v;
    atomicAdd(&denom[(size_t)d * heads + h], v);
  }
}

// Edge pass 3: out[dst] += alpha * h[src]. One wave per edge; lane strides
// the D columns. h and out both sit in L2 (51MB each < 192MB).
__global__ __launch_bounds__(256) void edge_agg_kernel(
    const int* __restrict__ ei, int E, int Nn, int heads, int C,
    const float* __restrict__ Hm, const float* __restrict__ ework,
    const float* __restrict__ denom, float* __restrict__ outacc) {
  const int lane = threadIdx.x & 31;
  int e = blockIdx.x * (blockDim.x >> 5) + (threadIdx.x >> 5);
  if (e >= E + Nn) return;
  int s = (e < E) ? ei[e] : (e - E);
  int d = (e < E) ? ei[E + e] : (e - E);
  const int D = heads * C;
  float al[4];
  for (int h = 0; h < heads; ++h)
    al[h] = ework[(size_t)e * heads + h] /
            (denom[(size_t)d * heads + h] + 1e-16f);
  const float* hrow = Hm + (size_t)s * D;
  float* orow = outacc + (size_t)d * D;
  for (int c = lane; c < D; c += 32)
    atomicAdd(&orow[c], al[c / C] * hrow[c]);
}

// ---------------------------------------------------------------------------
// out = relu(bn(acc + bias))
// ---------------------------------------------------------------------------
__global__ void bn_relu_kernel(const float* __restrict__ acc,
                               const float* __restrict__ bias,
                               const float* __restrict__ g,
                               const float* __restrict__ be,
                               const float* __restrict__ mn,
                               const float* __restrict__ vr,
                               float* __restrict__ out, size_t total, int D) {
  size_t i = (size_t)blockIdx.x * blockDim.x + threadIdx.x;
  size_t stride = (size_t)gridDim.x * blockDim.x;
  for (; i < total; i += stride) {
    int dcol = (int)(i % D);
    float x = acc[i] + bias[dcol];
    x = (x - mn[dcol]) * rsqrtf(vr[dcol] + BN_EPS) * g[dcol] + be[dcol];
    out[i] = fmaxf(x, 0.f);
  }
}

// ---------------------------------------------------------------------------
// Pooling: gate = relu(x4 @ pw1 + pb1) @ pw2 + pb2, then per-graph softmax.
// One wave per node; lane j owns hidden unit j (32 hidden units).
// ---------------------------------------------------------------------------
__global__ __launch_bounds__(256) void gate_kernel(
    const float* __restrict__ x4, const float* __restrict__ pw1,
    const float* __restrict__ pb1, const float* __restrict__ pw2,
    const float* __restrict__ pb2, const int* __restrict__ batch,
    float* __restrict__ gate, float* __restrict__ gm, int Nn) {
  const int lane = threadIdx.x & 31;
  int n = blockIdx.x * (blockDim.x >> 5) + (threadIdx.x >> 5);
  if (n >= Nn) return;
  const float* xr = x4 + (size_t)n * 128;
  float hsum = 0.f;
  for (int k = 0; k < 128; ++k) hsum += xr[k] * pw1[k * 32 + lane];
  float hv = fmaxf(hsum + pb1[lane], 0.f) * pw2[lane];
  for (int off = 16; off > 0; off >>= 1) hv += __shfl_down(hv, off, 32);
  if (lane == 0) {
    float gv = hv + pb2[0];
    gate[n] = gv;
    atomicMaxF(&gm[batch[n]], gv);
  }
}

__global__ void pool_sum_kernel(const float* __restrict__ gate,
                                const float* __restrict__ gm,
                                const int* __restrict__ batch,
                                float* __restrict__ ge,
                                float* __restrict__ gden, int Nn) {
  int n = blockIdx.x * blockDim.x + threadIdx.x;
  if (n >= Nn) return;
  int b = batch[n];
  float v = __expf(gate[n] - gm[b]);
  ge[n] = v;
  atomicAdd(&gden[b], v);
}

__global__ __launch_bounds__(256) void pool_emb_kernel(
    const float* __restrict__ x4, const float* __restrict__ ge,
    const float* __restrict__ gden, const int* __restrict__ batch,
    float* __restrict__ emb, int Nn) {
  const int lane = threadIdx.x & 31;
  int n = blockIdx.x * (blockDim.x >> 5) + (threadIdx.x >> 5);
  if (n >= Nn) return;
  int b = batch[n];
  float attn = ge[n] / (gden[b] + 1e-16f);
  const float* xr = x4 + (size_t)n * 128;
  float* er = emb + (size_t)b * 128;
  for (int c = lane; c < 128; c += 32) atomicAdd(&er[c], attn * xr[c]);
}

__global__ void final_kernel(const float* __restrict__ emb,
                             const float* __restrict__ gfeat,
                             const float* __restrict__ gw,
                             const float* __restrict__ gb,
                             float* __restrict__ out, int G) {
  int i = blockIdx.x * blockDim.x + threadIdx.x;
  if (i >= G * 160) return;
  int g = i / 160, c = i % 160;
  float v;
  if (c < 128) {
    v = emb[(size_t)g * 128 + c];
  } else {
    int j = c - 128;
    float s = gb[j];
    for (int k = 0; k < 7; ++k) s += gfeat[g * 7 + k] * gw[k * 32 + j];
    v = fmaxf(s, 0.f);
  }
  out[i] = v;
}

// ---------------------------------------------------------------------------
// Host side
// ---------------------------------------------------------------------------
struct LayerP {
  const float *W, *as, *ad, *b, *g, *be, *mn, *vr;
};

static void run_gat_layer(hipStream_t stream, const float* xin,
                          const float* xin2, int K, int heads, int C,
                          const LayerP& p, float* Apad, float* h, float* agg,
                          float* asb, float* adb, float* mmax, float* denom,
                          float* ework, const int* ei, int N, int E,
                          float* xout) {
  const int D = heads * C;
  const int Mpad = ((N + 127) / 128) * 128;
  const int Kpad = ((K + 63) / 64) * 64;

  // Pack (and fuse residual) into zero-padded A.
  size_t ptotal = (size_t)Mpad * Kpad;
  pack_kernel<<<(unsigned)((ptotal + 255) / 256), 256, 0, stream>>>(
      xin, xin2, Apad, N, K, Kpad, ptotal);

  dim3 gg(Mpad / 128, D / 64);
  gemm_wmma_kernel<<<gg, 256, 0, stream>>>(Apad, p.W, h, N, Kpad, K, D);
  alpha_kernel<<<N, 256, 0, stream>>>(h, p.as, p.ad, asb, adb, heads, C);

  size_t nh = (size_t)N * heads;
  fill_kernel<<<(unsigned)((nh + 255) / 256), 256, 0, stream>>>(mmax, -1e30f,
                                                                nh);
  fill_kernel<<<(unsigned)((nh + 255) / 256), 256, 0, stream>>>(denom, 0.f, nh);
  size_t nd = (size_t)N * D;
  fill_kernel<<<(unsigned)((nd + 255) / 256), 256, 0, stream>>>(agg, 0.f, nd);

  const int Etot = E + N;
  edge_max_kernel<<<(Etot + 255) / 256, 256, 0, stream>>>(ei, E, N, heads, asb,
                                                          adb, ework, mmax);
  edge_sum_kernel<<<(Etot + 255) / 256, 256, 0, stream>>>(ei, E, N, heads,
                                                          mmax, ework, denom);
  edge_agg_kernel<<<(Etot + 7) / 8, 256, 0, stream>>>(ei, E, N, heads, C, h,
                                                      ework, denom, agg);
  bn_relu_kernel<<<(unsigned)((nd + 255) / 256), 256, 0, stream>>>(
      agg, p.b, p.g, p.be, p.mn, p.vr, xout, nd, D);
}

extern "C" void kernel_launch(void* const* d_in, const int* in_sizes, int n_in,
                              void* d_out, int out_size, void* d_ws,
                              size_t ws_size, hipStream_t stream) {
  const float* x = (const float*)d_in[0];
  const int* ei = (const int*)d_in[1];
  const int* batch = (const int*)d_in[2];
  const float* gfeat = (const float*)d_in[3];
  const int N = in_sizes[2];       // batch: [N]
  const int E = in_sizes[1] / 2;   // edge_index: [2,E]
  const int G = in_sizes[3] / 7;   // global_features: [G,7]
  const int IN = in_sizes[0] / N;  // 58
  const int Etot = E + N;
  const int Mpad = ((N + 127) / 128) * 128;

  auto F = [&](int i) { return (const float*)d_in[i]; };
  LayerP L1{F(4), F(5), F(6), F(7), F(8), F(9), F(10), F(11)};
  LayerP L2{F(12), F(13), F(14), F(15), F(16), F(17), F(18), F(19)};
  LayerP L3{F(20), F(21), F(22), F(23), F(24), F(25), F(26), F(27)};
  LayerP L4{F(28), F(29), F(30), F(31), F(32), F(33), F(34), F(35)};
  const float* pw1 = F(36);
  const float* pb1 = F(37);
  const float* pw2 = F(38);
  const float* pb2 = F(39);
  const float* gw = F(40);
  const float* gb = F(41);

  // Workspace layout (~330MB of fp32)
  float* ws = (float*)d_ws;
  size_t off = 0;
  auto alloc = [&](size_t n) { float* p = ws + off; off += n; return p; };
  float* Apad = alloc((size_t)Mpad * 256);
  float* h = alloc((size_t)N * 256);
  float* agg = alloc((size_t)N * 256);
  float* x1 = alloc((size_t)N * 256);
  float* x2 = alloc((size_t)N * 256);
  float* x3 = alloc((size_t)N * 256);
  float* asb = alloc((size_t)N * 4);
  float* adb = alloc((size_t)N * 4);
  float* mmax = alloc((size_t)N * 4);
  float* denom = alloc((size_t)N * 4);
  float* ework = alloc((size_t)Etot * 4);
  float* gate = alloc((size_t)N);
  float* ge = alloc((size_t)N);
  float* gm = alloc((size_t)G);
  float* gden = alloc((size_t)G);
  float* emb = alloc((size_t)G * 128);
  float* x4 = x1;  // x1 is dead after layer-3 pack; reuse for x4 [N,128]

  // Layer 1: x[N,58] -> x1[N,256]
  run_gat_layer(stream, x, nullptr, IN, 4, 64, L1, Apad, h, agg, asb, adb,
                mmax, denom, ework, ei, N, E, x1);
  // Layer 2: x1 -> x2
  run_gat_layer(stream, x1, nullptr, 256, 4, 64, L2, Apad, h, agg, asb, adb,
                mmax, denom, ework, ei, N, E, x2);
  // Layer 3: (x1 + x2) -> x3   (residual add fused into pack)
  run_gat_layer(stream, x1, x2, 256, 4, 64, L3, Apad, h, agg, asb, adb, mmax,
                denom, ework, ei, N, E, x3);
  // Layer 4: x3 -> x4[N,128] (heads=1, C=128)
  run_gat_layer(stream, x3, nullptr, 256, 1, 128, L4, Apad, h, agg, asb, adb,
                mmax, denom, ework, ei, N, E, x4);

  // Global attention pooling
  fill_kernel<<<1, 256, 0, stream>>>(gm, -1e30f, (size_t)G);
  fill_kernel<<<1, 256, 0, stream>>>(gden, 0.f, (size_t)G);
  fill_kernel<<<(G * 128 + 255) / 256, 256, 0, stream>>>(emb, 0.f,
                                                         (size_t)G * 128);
  gate_kernel<<<(N + 7) / 8, 256, 0, stream>>>(x4, pw1, pb1, pw2, pb2, batch,
                                               gate, gm, N);
  pool_sum_kernel<<<(N + 255) / 256, 256, 0, stream>>>(gate, gm, batch, ge,
                                                       gden, N);
  pool_emb_kernel<<<(N + 7) / 8, 256, 0, stream>>>(x4, ge, gden, batch, emb, N);
  final_kernel<<<(G * 160 + 255) / 256, 256, 0, stream>>>(emb, gfeat, gw, gb,
                                                          (float*)d_out, G);
}